// HypAgg_35476429864969
// MI455X (gfx1250) — compile-verified
//
#include <hip/hip_runtime.h>
#include <math.h>

// ---------------------------------------------------------------------------
// HypAgg for MI455X (gfx1250, wave32).
// Sparse formulation: adj[r,c] = (s[r]+s[c])/2 for distinct edges (r,c);
// duplicates carry identical weights, so SET-scatter == dedup-once.
// WMMA (V_WMMA_F32_16X16X4_F32) computes the dense per-node stats:
//   s  = X * ones  (row sums, drives edge weights)
//   sq = diag(X X^T) (squared norms, drives logmap0)
// ---------------------------------------------------------------------------

typedef __attribute__((ext_vector_type(2))) float v2f;
typedef __attribute__((ext_vector_type(8))) float v8f;

#define DFEAT   128
#define MINN2   1e-30f        // MIN_NORM^2
#define BOUND   0.9999999f    // 1 - 1e-7 (artanh clip)
#define MAXNORM 0.996f        // (1 - 4e-3) / sqrt(c), c = 1

// --- Kernel 1: WMMA row stats. One wave per 16-row tile. --------------------
// f32 16x16x4 layouts (ISA 7.12.2):
//   A (16x4): lane m=l%16, vgpr0 holds K = 2*(l>=16), vgpr1 holds K+1
//   B (4x16): lane n=l%16, vgpr0 holds K = 2*(l>=16), vgpr1 holds K+1
//   C (16x16): vgpr j, lanes 0-15 -> M=j, N=l; lanes 16-31 -> M=8+j, N=l-16
// Passing the same register pair as A and B yields the Gram matrix of the tile.
__global__ void rowstats_wmma(const float* __restrict__ x,
                              float* __restrict__ s,
                              float* __restrict__ sq, int n) {
  const int lane  = threadIdx.x & 31;
  const int tile  = blockIdx.x * 8 + (threadIdx.x >> 5);
  const int base  = tile * 16;
  if (base >= n) return;                       // wave-uniform: EXEC stays all-1s
  const int m     = lane & 15;
  const int khalf = lane >> 4;                 // 0 or 1
  int row = base + m; if (row >= n) row = n - 1;   // clamp (no divergence)
  const float* px = x + (size_t)row * DFEAT + 2 * khalf;

  v8f cs = {};                                 // row-sum accumulator
  v8f cg = {};                                 // Gram accumulator
  v2f ones; ones[0] = 1.0f; ones[1] = 1.0f;
  for (int kk = 0; kk < DFEAT; kk += 4) {
    v2f a = *(const v2f*)(px + kk);
    cs = __builtin_amdgcn_wmma_f32_16x16x4_f32(false, a, false, ones, (short)0, cs, false, false);
    cg = __builtin_amdgcn_wmma_f32_16x16x4_f32(false, a, false, a,    (short)0, cg, false, false);
  }
  // Row sums: C[m][*] identical across N. Lane 0 -> M=0..7, lane 16 -> M=8..15.
  if ((lane & 15) == 0) {
    for (int j = 0; j < 8; ++j) {
      int r = base + 8 * khalf + j;
      if (r < n) s[r] = cs[j];
    }
  }
  // Gram diagonal C[M][M]: M<8 lives at lane M, vgpr M; M>=8 at lane 16+M-8... = M+16.
  if (lane < 8) {
    int r = base + lane;
    if (r < n) sq[r] = cg[lane];
  } else if (lane >= 24) {
    int r = base + (lane - 16);
    if (r < n) sq[r] = cg[lane - 24];
  }
}

// --- Kernel 2: xt = artanh(||x||)/||x|| * x ; acc = 0 -----------------------
__global__ void logmap_init(const float* __restrict__ x,
                            const float* __restrict__ sq,
                            float* __restrict__ xt,
                            float* __restrict__ acc, long total) {
  long i = (long)blockIdx.x * blockDim.x + threadIdx.x;
  if (i >= total) return;
  int r   = (int)(i >> 7);                     // / DFEAT
  float d  = sq[r];
  float pn = sqrtf(fmaxf(d, MINN2));
  float z  = fminf(pn, BOUND);
  float at = 0.5f * logf((1.0f + z) / (1.0f - z));   // artanh
  xt[i]  = (at / pn) * x[i];
  acc[i] = 0.0f;
}

// --- Kernel 3: clear dedup bitmap (re-run every call; graph-replay safe) ----
__global__ void zero_words(unsigned int* __restrict__ bm, long words) {
  long i      = (long)blockIdx.x * blockDim.x + threadIdx.x;
  long stride = (long)gridDim.x * blockDim.x;
  for (; i < words; i += stride) bm[i] = 0u;
}

// --- Kernel 4: dedup'd scatter SpMM. One wave per edge. ---------------------
__global__ void spmm_edges(const int* __restrict__ row,
                           const int* __restrict__ col,
                           const float* __restrict__ s,
                           const float* __restrict__ xt,
                           float* __restrict__ acc,
                           unsigned int* __restrict__ bm,
                           int n, int e) {
  const int lane = threadIdx.x & 31;
  const int eidx = blockIdx.x * 8 + (threadIdx.x >> 5);
  if (eidx >= e) return;
  const int r = row[eidx];
  const int c = col[eidx];

  int claimed = 0;
  if (lane == 0) {                             // exactly-once claim of (r,c)
    unsigned long long b = (unsigned long long)r * (unsigned)n + (unsigned)c;
    unsigned int bit = 1u << (unsigned)(b & 31);
    unsigned int old = atomicOr(&bm[b >> 5], bit);
    claimed = (old & bit) ? 0 : 1;
  }
  claimed = __shfl(claimed, 0);
  if (!claimed) return;                        // duplicate edge: same weight, drop

  const float  w = 0.5f * (s[r] + s[c]);
  const float4 v = ((const float4*)(xt + (size_t)c * DFEAT))[lane];
  float* dst = acc + (size_t)r * DFEAT + lane * 4;
  unsafeAtomicAdd(dst + 0, w * v.x);
  unsafeAtomicAdd(dst + 1, w * v.y);
  unsafeAtomicAdd(dst + 2, w * v.z);
  unsafeAtomicAdd(dst + 3, w * v.w);
}

// --- Kernel 5: out = proj(expmap0(acc)). One wave per row. ------------------
__global__ void finalize(const float* __restrict__ acc,
                         float* __restrict__ out, int n) {
  const int lane = threadIdx.x & 31;
  const int r    = blockIdx.x * 8 + (threadIdx.x >> 5);
  if (r >= n) return;
  const float4 u = ((const float4*)(acc + (size_t)r * DFEAT))[lane];
  float p = u.x * u.x + u.y * u.y + u.z * u.z + u.w * u.w;
  for (int off = 16; off; off >>= 1) p += __shfl_xor(p, off);
  const float un = sqrtf(fmaxf(p, MINN2));
  float sc = tanhf(un) / un;                             // expmap0 scale
  const float yn = sqrtf(fmaxf(sc * sc * p, MINN2));     // ||expmap0(u)||
  if (yn > MAXNORM) sc *= MAXNORM / yn;                  // proj
  float4 y;
  y.x = sc * u.x; y.y = sc * u.y; y.z = sc * u.z; y.w = sc * u.w;
  ((float4*)(out + (size_t)r * DFEAT))[lane] = y;
}

static inline size_t align256(size_t b) { return (b + 255) & ~(size_t)255; }

extern "C" void kernel_launch(void* const* d_in, const int* in_sizes, int n_in,
                              void* d_out, int out_size, void* d_ws, size_t ws_size,
                              hipStream_t stream) {
  const float* x  = (const float*)d_in[0];
  const int*   ei = (const int*)d_in[1];
  const int n = in_sizes[0] / DFEAT;           // 10000
  const int e = in_sizes[1] / 2;               // 320000
  const int* row = ei;
  const int* col = ei + e;

  // Workspace layout (~23 MB total for N=10000): s, sq, xt, acc, bitmap.
  char* ws = (char*)d_ws;
  size_t off = 0;
  float* s   = (float*)(ws + off); off += align256((size_t)n * 4);
  float* sq  = (float*)(ws + off); off += align256((size_t)n * 4);
  float* xt  = (float*)(ws + off); off += align256((size_t)n * DFEAT * 4);
  float* acc = (float*)(ws + off); off += align256((size_t)n * DFEAT * 4);
  unsigned int* bm = (unsigned int*)(ws + off);
  const long words = ((long)n * (long)n + 31) / 32;

  const int tiles = (n + 15) / 16;
  rowstats_wmma<<<(tiles + 7) / 8, 256, 0, stream>>>(x, s, sq, n);

  const long total = (long)n * DFEAT;
  logmap_init<<<(int)((total + 255) / 256), 256, 0, stream>>>(x, sq, xt, acc, total);

  zero_words<<<2048, 256, 0, stream>>>(bm, words);

  spmm_edges<<<(e + 7) / 8, 256, 0, stream>>>(row, col, s, xt, acc, bm, n, e);

  finalize<<<(n + 7) / 8, 256, 0, stream>>>(acc, (float*)d_out, n);
}